// MaternKernel_37684043055329
// MI455X (gfx1250) — compile-verified
//
#include <hip/hip_runtime.h>

// Matern-1.5 kernel matrix: out[i,j] = (1 + sqrt3*d) * exp(-sqrt3*d),
// d = ||t1[i] - t2[j]||.  (Mean-centering cancels: t1_s - t2_s = t1 - t2;
// the 1e-15 distance clip is a value-level no-op in f32 and is elided.)
//
// WMMA strategy: dist^2 computed ENTIRELY in the accumulator:
//   wmma#1: A'={n1[M],1,0,0}, B'={1,n2[N],0,0}  ->  acc = n1[M] + n2[N]
//   wmma#2: (-2*A_k0..3) x B_k0..3              ->  acc -= 2*dot(lo)
//   wmma#3: (-2*A_k4..7) x B_k4..7              ->  acc -= 2*dot(hi) = dist^2
// Epilogue per element: v_med3 clamp, v_sqrt, v_fma prefactor, v_mul, v_exp,
// v_mul, immediate-offset nontemporal store.

typedef __attribute__((ext_vector_type(2))) float v2f;
typedef __attribute__((ext_vector_type(4))) float v4f;
typedef __attribute__((ext_vector_type(8))) float v8f;

#define DIMS 8
#define WAVES_PER_BLOCK 8

// MC > 0: compile-time column count (row stride folds into store IOFFSET).
// MC == 0: runtime column count (fallback).
template <int MC>
__global__ __launch_bounds__(256) void matern_wmma_kernel(
    const float* __restrict__ t1, const float* __restrict__ t2,
    float* __restrict__ out, int N, int Mrt) {
  const int M = (MC > 0) ? MC : Mrt;

  const int lane = threadIdx.x & 31;
  const int wave = threadIdx.x >> 5;
  const int half = lane >> 4;   // 0: lanes 0-15, 1: lanes 16-31
  const int lmod = lane & 15;

  const int tile_i = blockIdx.y;                          // row tile
  const int tile_j = blockIdx.x * WAVES_PER_BLOCK + wave; // col tile
  const int row0 = tile_i << 4;
  const int col0 = tile_j << 4;
  if (col0 >= M || row0 >= N) return;  // wave-uniform scalar branch (EXEC stays all-1s)

  // Each lane owns one full row of each 16-row tile (lanes 16-31 duplicate).
  const float* arow = t1 + (size_t)(unsigned)((row0 + lmod) * DIMS);
  const float* brow = t2 + (size_t)(unsigned)((col0 + lmod) * DIMS);

  v4f a_lo = *(const v4f*)(arow);       // K = 0..3
  v4f a_hi = *(const v4f*)(arow + 4);   // K = 4..7
  v4f b_lo = *(const v4f*)(brow);
  v4f b_hi = *(const v4f*)(brow + 4);

  // Row norms (this lane's row of each tile).
  float nn1 = a_lo.x * a_lo.x + a_lo.y * a_lo.y + a_lo.z * a_lo.z + a_lo.w * a_lo.w
            + a_hi.x * a_hi.x + a_hi.y * a_hi.y + a_hi.z * a_hi.z + a_hi.w * a_hi.w;
  float nn2 = b_lo.x * b_lo.x + b_lo.y * b_lo.y + b_lo.z * b_lo.z + b_lo.w * b_lo.w
            + b_hi.x * b_hi.x + b_hi.y * b_hi.y + b_hi.z * b_hi.z + b_hi.w * b_hi.w;

  // WMMA fragments per ISA 16x4 f32 layout:
  //   VGPR0 holds K = 2*half, VGPR1 holds K = 2*half + 1  (lane -> M resp. N).
  // A fragments pre-scaled by -2 so the dot WMMAs accumulate -2*<a,b>.
  v2f a0 = (half ? a_lo.zw : a_lo.xy) * -2.0f;   // K = {0,1} | {2,3}
  v2f a1 = (half ? a_hi.zw : a_hi.xy) * -2.0f;   // K = {4,5} | {6,7}
  v2f b0 = half ? b_lo.zw : b_lo.xy;
  v2f b1 = half ? b_hi.zw : b_hi.xy;

  // Norm-injection fragments: A'[M][:] = {n1[M], 1, 0, 0}, B'[:][N] = {1, n2[N], 0, 0}
  v2f ap = half ? (v2f){0.0f, 0.0f} : (v2f){nn1, 1.0f};
  v2f bp = half ? (v2f){0.0f, 0.0f} : (v2f){1.0f, nn2};

  v8f acc = {};
  acc = __builtin_amdgcn_wmma_f32_16x16x4_f32(false, ap, false, bp,
                                              (short)0, acc, false, false);
  acc = __builtin_amdgcn_wmma_f32_16x16x4_f32(false, a0, false, b0,
                                              (short)0, acc, false, false);
  acc = __builtin_amdgcn_wmma_f32_16x16x4_f32(false, a1, false, b1,
                                              (short)0, acc, false, false);
  // acc[r] = dist^2 for element (M = r + 8*half, N = lmod)

  const float sqrt3 = 1.7320508075688772f;
  const float nc    = -2.4988856526260397f;  // -sqrt3 * log2(e): exp(-sqrt3*d) = exp2(nc*d)
  const float inf   = __builtin_inff();
  float* orow = out + (size_t)(unsigned)((row0 + 8 * half) * (unsigned)M)
                    + (unsigned)(col0 + lmod);
#pragma unroll
  for (int r = 0; r < 8; ++r) {
    float d2   = __builtin_amdgcn_fmed3f(acc[r], 0.0f, inf);  // clamp to [0, inf)
    float dist = __builtin_amdgcn_sqrtf(d2);
    float val  = __builtin_fmaf(sqrt3, dist, 1.0f) *
                 __builtin_amdgcn_exp2f(nc * dist);
    __builtin_nontemporal_store(val, orow + (size_t)((unsigned)r * (unsigned)M));
  }
}

extern "C" void kernel_launch(void* const* d_in, const int* in_sizes, int n_in,
                              void* d_out, int out_size, void* d_ws, size_t ws_size,
                              hipStream_t stream) {
  const float* t1 = (const float*)d_in[0];
  const float* t2 = (const float*)d_in[1];
  float* out = (float*)d_out;
  const int N = in_sizes[0] / DIMS;  // 8192 rows of t1
  const int M = in_sizes[1] / DIMS;  // 8192 rows of t2

  const int tiles_i = N / 16;                                            // 512
  const int tiles_j = (M / 16 + WAVES_PER_BLOCK - 1) / WAVES_PER_BLOCK;  // 64
  dim3 grid(tiles_j, tiles_i);
  dim3 block(32 * WAVES_PER_BLOCK);  // 8 wave32 waves

  if (M == 8192) {
    matern_wmma_kernel<8192><<<grid, block, 0, stream>>>(t1, t2, out, N, M);
  } else {
    matern_wmma_kernel<0><<<grid, block, 0, stream>>>(t1, t2, out, N, M);
  }
}